// NewtonIteration_65609920413788
// MI455X (gfx1250) — compile-verified
//
#include <hip/hip_runtime.h>

// ---------------------------------------------------------------------------
// Glacier hydrology update on an unstructured mesh (1M nodes, 2M links).
// MI455X (gfx1250, wave32) tuning:
//   - b128 vector loads on all streaming arrays (4 elements / thread)
//   - fp32 scatter-adds via hardware global_atomic_add_f32 @ SCOPE_DEV
//   - random per-link gathers are L2-resident (working set << 192MB L2)
//   - 256-thread blocks = 8 wave32 waves
// Roofline: ~90MB of traffic @ 23.3TB/s, trivially low FLOPs -> memory bound.
// No dense-matrix structure exists in this workload, so WMMA is inapplicable.
// ---------------------------------------------------------------------------

#define TPB 256

__device__ __forceinline__ void fadd_atomic(float* p, float v) {
    // Hardware fp32 atomic (global_atomic_add_f32 scope:SCOPE_DEV), no CAS loop.
    unsafeAtomicAdd(p, v);
}

// ---------------- node physics (shared by vector + tail kernels) ------------
__device__ __forceinline__ void node_physics(float head_v, float bed, float P0,
                                             float geo, unsigned char bnd,
                                             float usum, float deg,
                                             float& h_out, float& cond_out,
                                             float& mc_out) {
    float h  = bnd ? bed : head_v;
    float wp = (1000.0f * 9.81f) * (h - bed);     // rho_w * g * (h - b)
    float N  = P0 - wp;
    N = (N > P0)   ? P0   : N;                    // clamp above by overburden
    N = (N < 1e4f) ? 1e4f : N;                    // MIN_EFFECTIVE_PRESSURE

    float u_node    = usum / fmaxf(deg, 1.0f);
    float friction  = fabsf(u_node * (0.6f * N));           // TILL_FRICTION
    float melt_flux = (geo + friction) / 3.34e5f;           // LATENT_HEAT

    float N3   = N * N * N;
    float cond = melt_flux / 917.0f / (6e-24f * N3);        // rho_i, fluidity

    const float melt_k = (float)(1.0 / 1000.0 - 1.0 / 917.0);
    mc_out   = melt_flux * melt_k + 6e-24f * N3 * cond;     // melt + closure
    h_out    = h;
    cond_out = cond;
}

// ---------------- link flux physics -----------------------------------------
__device__ __forceinline__ float link_flux_val(float hT, float hH, float cT,
                                               float cH, float Re, float L) {
    float grad  = (hH - hT) / L;
    float cl    = 0.5f * (cT + cH);
    float denom = (float)(12.0 * 1.787e-3) * (1.0f + 1e-3f * Re);
    float T     = cl * cl * cl * 9.81f / denom;
    return -T * grad * L;     // keep the grad*L round-trip like the reference
}

// ======================= Pass 1: link accumulation ==========================
__global__ void link_accum4(const int4* __restrict__ tail,
                            const int4* __restrict__ head,
                            const float4* __restrict__ u_slide,
                            float* __restrict__ u_sum,
                            float* __restrict__ degree,
                            int n_vec) {
    int i = blockIdx.x * TPB + threadIdx.x;
    if (i >= n_vec) return;
    int4   t = tail[i];
    int4   h = head[i];
    float4 u = u_slide[i];
    fadd_atomic(&u_sum[t.x], u.x);  fadd_atomic(&u_sum[h.x], u.x);
    fadd_atomic(&u_sum[t.y], u.y);  fadd_atomic(&u_sum[h.y], u.y);
    fadd_atomic(&u_sum[t.z], u.z);  fadd_atomic(&u_sum[h.z], u.z);
    fadd_atomic(&u_sum[t.w], u.w);  fadd_atomic(&u_sum[h.w], u.w);
    fadd_atomic(&degree[t.x], 1.0f); fadd_atomic(&degree[h.x], 1.0f);
    fadd_atomic(&degree[t.y], 1.0f); fadd_atomic(&degree[h.y], 1.0f);
    fadd_atomic(&degree[t.z], 1.0f); fadd_atomic(&degree[h.z], 1.0f);
    fadd_atomic(&degree[t.w], 1.0f); fadd_atomic(&degree[h.w], 1.0f);
}

__global__ void link_accum_tail(const int* __restrict__ tail,
                                const int* __restrict__ head,
                                const float* __restrict__ u_slide,
                                float* __restrict__ u_sum,
                                float* __restrict__ degree,
                                int start, int n_links) {
    int i = start + blockIdx.x * TPB + threadIdx.x;
    if (i >= n_links) return;
    int t = tail[i], h = head[i];
    float u = u_slide[i];
    fadd_atomic(&u_sum[t], u);     fadd_atomic(&u_sum[h], u);
    fadd_atomic(&degree[t], 1.0f); fadd_atomic(&degree[h], 1.0f);
}

// ======================= Pass 2: node physics ===============================
__global__ void node_pass4(const float4* __restrict__ head,
                           const float4* __restrict__ bedrock,
                           const float4* __restrict__ overburden,
                           const float4* __restrict__ geothermal,
                           const uchar4* __restrict__ is_bnd,
                           const float4* __restrict__ u_sum,
                           const float4* __restrict__ degree,
                           float4* __restrict__ h_bc,
                           float4* __restrict__ conduit,
                           float4* __restrict__ melt_closure,  // -> d_out
                           int n_vec) {
    int i = blockIdx.x * TPB + threadIdx.x;
    if (i >= n_vec) return;
    float4 hd = head[i], bd = bedrock[i], P0 = overburden[i], ge = geothermal[i];
    float4 us = u_sum[i], dg = degree[i];
    uchar4 bn = is_bnd[i];
    float4 ho, co, mc;
    node_physics(hd.x, bd.x, P0.x, ge.x, bn.x, us.x, dg.x, ho.x, co.x, mc.x);
    node_physics(hd.y, bd.y, P0.y, ge.y, bn.y, us.y, dg.y, ho.y, co.y, mc.y);
    node_physics(hd.z, bd.z, P0.z, ge.z, bn.z, us.z, dg.z, ho.z, co.z, mc.z);
    node_physics(hd.w, bd.w, P0.w, ge.w, bn.w, us.w, dg.w, ho.w, co.w, mc.w);
    h_bc[i] = ho; conduit[i] = co; melt_closure[i] = mc;
}

__global__ void node_pass_tail(const float* __restrict__ head,
                               const float* __restrict__ bedrock,
                               const float* __restrict__ overburden,
                               const float* __restrict__ geothermal,
                               const unsigned char* __restrict__ is_bnd,
                               const float* __restrict__ u_sum,
                               const float* __restrict__ degree,
                               float* __restrict__ h_bc,
                               float* __restrict__ conduit,
                               float* __restrict__ melt_closure,
                               int start, int n_nodes) {
    int i = start + blockIdx.x * TPB + threadIdx.x;
    if (i >= n_nodes) return;
    float ho, co, mc;
    node_physics(head[i], bedrock[i], overburden[i], geothermal[i], is_bnd[i],
                 u_sum[i], degree[i], ho, co, mc);
    h_bc[i] = ho; conduit[i] = co; melt_closure[i] = mc;
}

// ======================= Pass 3: link fluxes ================================
__global__ void link_flux4(const int4* __restrict__ tail,
                           const int4* __restrict__ head,
                           const float4* __restrict__ Re,
                           const float4* __restrict__ length,
                           const float* __restrict__ h_bc,
                           const float* __restrict__ conduit,
                           float* __restrict__ net_flux,
                           int n_vec) {
    int i = blockIdx.x * TPB + threadIdx.x;
    if (i >= n_vec) return;
    int4   t = tail[i], h = head[i];
    float4 r = Re[i],   L = length[i];

    float fx = link_flux_val(h_bc[t.x], h_bc[h.x], conduit[t.x], conduit[h.x], r.x, L.x);
    float fy = link_flux_val(h_bc[t.y], h_bc[h.y], conduit[t.y], conduit[h.y], r.y, L.y);
    float fz = link_flux_val(h_bc[t.z], h_bc[h.z], conduit[t.z], conduit[h.z], r.z, L.z);
    float fw = link_flux_val(h_bc[t.w], h_bc[h.w], conduit[t.w], conduit[h.w], r.w, L.w);

    fadd_atomic(&net_flux[t.x],  fx);  fadd_atomic(&net_flux[h.x], -fx);
    fadd_atomic(&net_flux[t.y],  fy);  fadd_atomic(&net_flux[h.y], -fy);
    fadd_atomic(&net_flux[t.z],  fz);  fadd_atomic(&net_flux[h.z], -fz);
    fadd_atomic(&net_flux[t.w],  fw);  fadd_atomic(&net_flux[h.w], -fw);
}

__global__ void link_flux_tail(const int* __restrict__ tail,
                               const int* __restrict__ head,
                               const float* __restrict__ Re,
                               const float* __restrict__ length,
                               const float* __restrict__ h_bc,
                               const float* __restrict__ conduit,
                               float* __restrict__ net_flux,
                               int start, int n_links) {
    int i = start + blockIdx.x * TPB + threadIdx.x;
    if (i >= n_links) return;
    int t = tail[i], h = head[i];
    float f = link_flux_val(h_bc[t], h_bc[h], conduit[t], conduit[h], Re[i], length[i]);
    fadd_atomic(&net_flux[t],  f);
    fadd_atomic(&net_flux[h], -f);
}

// ======================= Pass 4: finalize ===================================
__global__ void finalize4(const float4* __restrict__ net_flux,
                          const float4* __restrict__ area,
                          const uchar4* __restrict__ is_bnd,
                          float4* __restrict__ out,   // holds melt+closure on entry
                          int n_vec) {
    int i = blockIdx.x * TPB + threadIdx.x;
    if (i >= n_vec) return;
    float4 nf = net_flux[i], ar = area[i], mc = out[i];
    uchar4 bn = is_bnd[i];
    float4 r;
    r.x = (bn.x ? 0.0f : nf.x / ar.x) - mc.x;
    r.y = (bn.y ? 0.0f : nf.y / ar.y) - mc.y;
    r.z = (bn.z ? 0.0f : nf.z / ar.z) - mc.z;
    r.w = (bn.w ? 0.0f : nf.w / ar.w) - mc.w;
    out[i] = r;
}

__global__ void finalize_tail(const float* __restrict__ net_flux,
                              const float* __restrict__ area,
                              const unsigned char* __restrict__ is_bnd,
                              float* __restrict__ out,
                              int start, int n_nodes) {
    int i = start + blockIdx.x * TPB + threadIdx.x;
    if (i >= n_nodes) return;
    float mc = out[i];
    out[i] = (is_bnd[i] ? 0.0f : net_flux[i] / area[i]) - mc;
}

// ============================================================================
extern "C" void kernel_launch(void* const* d_in, const int* in_sizes, int n_in,
                              void* d_out, int out_size, void* d_ws, size_t ws_size,
                              hipStream_t stream) {
    const float* head       = (const float*)d_in[0];          // [N]
    const float* Re         = (const float*)d_in[1];          // [L]
    const float* length     = (const float*)d_in[2];          // [L]
    const float* area       = (const float*)d_in[3];          // [N]
    const float* bedrock    = (const float*)d_in[4];          // [N]
    const float* overburden = (const float*)d_in[5];          // [N]
    const float* geothermal = (const float*)d_in[6];          // [N]
    const float* u_slide    = (const float*)d_in[7];          // [L]
    const int*   link_tail  = (const int*)d_in[8];            // [L]
    const int*   link_head  = (const int*)d_in[9];            // [L]
    const unsigned char* is_bnd = (const unsigned char*)d_in[10]; // [N] bool

    const int n_nodes = in_sizes[0];
    const int n_links = in_sizes[1];

    // Workspace layout (floats): [u_sum | degree | net_flux | h_bc | conduit]
    float* ws       = (float*)d_ws;
    float* u_sum    = ws + (size_t)0 * n_nodes;
    float* degree   = ws + (size_t)1 * n_nodes;
    float* net_flux = ws + (size_t)2 * n_nodes;
    float* h_bc     = ws + (size_t)3 * n_nodes;
    float* conduit  = ws + (size_t)4 * n_nodes;

    float* out = (float*)d_out;

    // Zero the three atomic accumulators every call (graph-capture-safe).
    hipMemsetAsync(d_ws, 0, (size_t)3 * n_nodes * sizeof(float), stream);

    const int lv = n_links / 4, lr = n_links - lv * 4;   // link vec + remainder
    const int nv = n_nodes / 4, nr = n_nodes - nv * 4;   // node vec + remainder
    dim3 blk(TPB);
    dim3 gl((lv + TPB - 1) / TPB);
    dim3 gn((nv + TPB - 1) / TPB);

    // ---- Pass 1: degree + u_slide sums --------------------------------------
    link_accum4<<<gl, blk, 0, stream>>>((const int4*)link_tail, (const int4*)link_head,
                                        (const float4*)u_slide, u_sum, degree, lv);
    if (lr) link_accum_tail<<<1, TPB, 0, stream>>>(link_tail, link_head, u_slide,
                                                   u_sum, degree, lv * 4, n_links);

    // ---- Pass 2: node physics ----------------------------------------------
    node_pass4<<<gn, blk, 0, stream>>>((const float4*)head, (const float4*)bedrock,
                                       (const float4*)overburden, (const float4*)geothermal,
                                       (const uchar4*)is_bnd, (const float4*)u_sum,
                                       (const float4*)degree, (float4*)h_bc,
                                       (float4*)conduit, (float4*)out, nv);
    if (nr) node_pass_tail<<<1, TPB, 0, stream>>>(head, bedrock, overburden, geothermal,
                                                  is_bnd, u_sum, degree, h_bc, conduit,
                                                  out, nv * 4, n_nodes);

    // ---- Pass 3: link fluxes -> net divergence -----------------------------
    link_flux4<<<gl, blk, 0, stream>>>((const int4*)link_tail, (const int4*)link_head,
                                       (const float4*)Re, (const float4*)length,
                                       h_bc, conduit, net_flux, lv);
    if (lr) link_flux_tail<<<1, TPB, 0, stream>>>(link_tail, link_head, Re, length,
                                                  h_bc, conduit, net_flux, lv * 4, n_links);

    // ---- Pass 4: finalize ---------------------------------------------------
    finalize4<<<gn, blk, 0, stream>>>((const float4*)net_flux, (const float4*)area,
                                      (const uchar4*)is_bnd, (float4*)out, nv);
    if (nr) finalize_tail<<<1, TPB, 0, stream>>>(net_flux, area, is_bnd, out,
                                                 nv * 4, n_nodes);
}